// GraphSAGEClassfication_86053964743053
// MI455X (gfx1250) — compile-verified
//
#include <hip/hip_runtime.h>

typedef __attribute__((ext_vector_type(2))) float v2f;
typedef __attribute__((ext_vector_type(8))) float v8f;

#define DH 64

// ---------------------------------------------------------------- utilities
__global__ void zero_kernel(float* __restrict__ p, long n) {
    long i = (long)blockIdx.x * blockDim.x + threadIdx.x;
    long stride = (long)gridDim.x * blockDim.x;
    for (; i < n; i += stride) p[i] = 0.0f;
}

__global__ void degree_kernel(const int* __restrict__ dst, float* __restrict__ cnt, int E) {
    int i = blockIdx.x * blockDim.x + threadIdx.x;
    int stride = gridDim.x * blockDim.x;
    for (; i < E; i += stride) atomicAdd(&cnt[dst[i]], 1.0f);
}

__global__ void inv_kernel(const float* __restrict__ cnt, float* __restrict__ inv, int N) {
    int i = blockIdx.x * blockDim.x + threadIdx.x;
    if (i < N) inv[i] = 1.0f / fmaxf(cnt[i], 1.0f);
}

// Mean-aggregate scatter: one float per work item; 64 consecutive lanes cover one
// edge's 256B feature row -> coalesced gather + coalesced atomic targets.
__global__ void scatter_kernel(const float* __restrict__ x, const int* __restrict__ src,
                               const int* __restrict__ dst, float* __restrict__ agg,
                               long total /* E*64 */) {
    long i = (long)blockIdx.x * blockDim.x + threadIdx.x;
    long stride = (long)gridDim.x * blockDim.x;
    for (; i < total; i += stride) {
        int e = (int)(i >> 6);
        int c = (int)(i & 63);
        atomicAdd(&agg[(long)dst[e] * DH + c], x[(long)src[e] * DH + c]);
    }
}

// ---------------------------------------------------------------- WMMA GEMM
// DUAL:  out = act( (A1 * invdeg[row]) @ W1 + A2 @ W2 + bias )
// !DUAL: out = act( A1 @ W1 + bias )
// All f32, K = DH = 64, V_WMMA_F32_16X16X4_F32.
// One wave computes a full 16x64 output row-tile (4 x 16x16 accumulators) so the
// A fragment is loaded once and fed to 4 independent WMMA chains per k-step.
// A layout (16x4 f32): lanes 0-15 hold M=lane {K=k0,k0+1}; lanes 16-31 {K=k0+2,k0+3}.
// B layout (4x16 f32): lanes 0-15 hold N=lane {K=k0,k0+1}; lanes 16-31 {K=k0+2,k0+3}.
// C/D (16x16 f32): VGPR r -> M = r + 8*(lane/16), N = lane%16.
template <bool DUAL, bool RELU>
__global__ __launch_bounds__(256)
void gemm_wmma(const float* __restrict__ A1, const float* __restrict__ invdeg,
               const float* __restrict__ W1,
               const float* __restrict__ A2, const float* __restrict__ W2,
               const float* __restrict__ bias, float* __restrict__ out,
               int row_tiles) {
    int wave = (int)((blockIdx.x * blockDim.x + threadIdx.x) >> 5);
    int lane = threadIdx.x & 31;
    if (wave >= row_tiles) return;     // wave-uniform; EXEC all-ones below

    int lo = lane & 15;
    int hi = lane >> 4;                // 0 or 1
    int row = wave * 16 + lo;          // M index owned by this lane (A side)

    float s1 = DUAL ? invdeg[row] : 1.0f;

    const float* a1p = A1 + (long)row * DH + 2 * hi;
    const float* a2p = DUAL ? (A2 + (long)row * DH + 2 * hi) : nullptr;

    v8f c[4];
    #pragma unroll
    for (int t = 0; t < 4; ++t) c[t] = (v8f){0.f, 0.f, 0.f, 0.f, 0.f, 0.f, 0.f, 0.f};

    #pragma unroll 4
    for (int k0 = 0; k0 < DH; k0 += 4) {
        int kr = k0 + 2 * hi;
        v2f a1 = *(const v2f*)(a1p + k0);
        if (DUAL) { a1.x *= s1; a1.y *= s1; }
        v2f a2;
        if (DUAL) a2 = *(const v2f*)(a2p + k0);

        const float* w1p = W1 + kr * DH + lo;          // row kr, +256B for row kr+1
        const float* w2p = DUAL ? (W2 + kr * DH + lo) : nullptr;

        #pragma unroll
        for (int t = 0; t < 4; ++t) {
            v2f b1;
            b1.x = w1p[t * 16];
            b1.y = w1p[DH + t * 16];
            c[t] = __builtin_amdgcn_wmma_f32_16x16x4_f32(false, a1, false, b1, (short)0,
                                                         c[t], false, false);
            if (DUAL) {
                v2f b2;
                b2.x = w2p[t * 16];
                b2.y = w2p[DH + t * 16];
                c[t] = __builtin_amdgcn_wmma_f32_16x16x4_f32(false, a2, false, b2,
                                                             (short)0, c[t], false, false);
            }
        }
    }

    float* orow = out + (long)(wave * 16) * DH;
    #pragma unroll
    for (int t = 0; t < 4; ++t) {
        int colt = t * 16 + lo;
        float bb = bias[colt];
        #pragma unroll
        for (int r = 0; r < 8; ++r) {
            float v = c[t][r] + bb;
            if (RELU) v = fmaxf(v, 0.0f);
            int m = r + 8 * hi;
            orow[(long)m * DH + colt] = v;   // half-wave writes 64B contiguous
        }
    }
}

// ---------------------------------------------------------------- head: 64->10 + log_softmax
__global__ void head_kernel(const float* __restrict__ h, const float* __restrict__ W,
                            const float* __restrict__ b, float* __restrict__ out, int N) {
    int i = blockIdx.x * blockDim.x + threadIdx.x;
    if (i >= N) return;
    const float* hp = h + (long)i * DH;
    float logit[10];
    #pragma unroll
    for (int c = 0; c < 10; ++c) logit[c] = b[c];
    for (int k = 0; k < DH; ++k) {
        float hv = hp[k];
        #pragma unroll
        for (int c = 0; c < 10; ++c) logit[c] = fmaf(hv, W[k * 10 + c], logit[c]);
    }
    float m = logit[0];
    #pragma unroll
    for (int c = 1; c < 10; ++c) m = fmaxf(m, logit[c]);
    float s = 0.0f;
    #pragma unroll
    for (int c = 0; c < 10; ++c) s += __expf(logit[c] - m);
    float lse = m + __logf(s);
    float* op = out + (long)i * 10;
    #pragma unroll
    for (int c = 0; c < 10; ++c) op[c] = logit[c] - lse;
}

// ---------------------------------------------------------------- launch
extern "C" void kernel_launch(void* const* d_in, const int* in_sizes, int n_in,
                              void* d_out, int out_size, void* d_ws, size_t ws_size,
                              hipStream_t stream) {
    const float* x     = (const float*)d_in[0];
    const int*   ei    = (const int*)d_in[1];   // (2, E) int32 (JAX x64 disabled)
    const float* Wl1   = (const float*)d_in[2];
    const float* Wr1   = (const float*)d_in[3];
    const float* b1    = (const float*)d_in[4];
    const float* Wl2   = (const float*)d_in[5];
    const float* Wr2   = (const float*)d_in[6];
    const float* b2    = (const float*)d_in[7];
    const float* Wlin1 = (const float*)d_in[8];
    const float* blin1 = (const float*)d_in[9];
    const float* Wlin2 = (const float*)d_in[10];
    const float* blin2 = (const float*)d_in[11];

    const int N = in_sizes[0] / DH;   // 100000
    const int E = in_sizes[1] / 2;    // 1000000
    const int* src = ei;
    const int* dst = ei + E;

    // workspace layout (floats): cnt[N] | inv[N] | agg[N*64] | h1[N*64] | h2[N*64]
    float* ws  = (float*)d_ws;
    float* cnt = ws;
    float* inv = cnt + N;
    float* agg = inv + N;
    float* h1  = agg + (size_t)N * DH;
    float* h2  = h1 + (size_t)N * DH;
    float* h3  = agg;                 // agg buffer dead after layer-2 GEMM -> reuse

    const int TB = 256;
    const int row_tiles = N / 16;                    // 6250 (N multiple of 16)
    const int gemm_blocks = (row_tiles + 7) / 8;     // 1 wave per 16x64 row tile

    // degrees (shared by both layers)
    zero_kernel<<<1024, TB, 0, stream>>>(cnt, (long)N);
    zero_kernel<<<4096, TB, 0, stream>>>(agg, (long)N * DH);
    degree_kernel<<<1024, TB, 0, stream>>>(dst, cnt, E);
    inv_kernel<<<(N + TB - 1) / TB, TB, 0, stream>>>(cnt, inv, N);

    // layer 1
    scatter_kernel<<<8192, TB, 0, stream>>>(x, src, dst, agg, (long)E * DH);
    gemm_wmma<true, true><<<gemm_blocks, TB, 0, stream>>>(agg, inv, Wl1, x, Wr1, b1, h1,
                                                          row_tiles);
    // layer 2
    zero_kernel<<<4096, TB, 0, stream>>>(agg, (long)N * DH);
    scatter_kernel<<<8192, TB, 0, stream>>>(h1, src, dst, agg, (long)E * DH);
    gemm_wmma<true, true><<<gemm_blocks, TB, 0, stream>>>(agg, inv, Wl2, h1, Wr2, b2, h2,
                                                          row_tiles);
    // lin1 + relu
    gemm_wmma<false, true><<<gemm_blocks, TB, 0, stream>>>(h2, nullptr, Wlin1, nullptr,
                                                           nullptr, blin1, h3, row_tiles);
    // lin2 + log_softmax
    head_kernel<<<(N + TB - 1) / TB, TB, 0, stream>>>(h3, Wlin2, blin2, (float*)d_out, N);
}